// CosformerCausalAttention_56367150793410
// MI455X (gfx1250) — compile-verified
//
#include <hip/hip_runtime.h>
#include <math.h>
#include <stdint.h>

// ---------------------------------------------------------------------------
// CosFormer causal attention for MI455X (gfx1250, wave32, WMMA f32).
//   B=2, T=1024, C=512, H=8, d=64
//   Pass 1 : qkv = x @ w_qkv^T + b_qkv      (WMMA f32 GEMM, async-LDS B tile)
//   Pass 2a: per-(b,h,chunk) outer-product totals
//   Pass 2b: chunked causal scan (seq depth 1024 -> 128)
//   Pass 3 : out = attn @ w_out^T + b_out   (same WMMA GEMM)
// ---------------------------------------------------------------------------

typedef __attribute__((ext_vector_type(2))) float v2f;
typedef __attribute__((ext_vector_type(8))) float v8f;

#define BB 2
#define TT 1024
#define CC 512
#define HH 8
#define DD 64
#define MM (BB * TT)
#define EPS 1e-6f

#define NCHUNK 8
#define CHUNK (TT / NCHUNK)
#define PART_STRIDE (2 * DD * DD + 2 * DD)   // 8320 floats per (bh,chunk)

// GEMM tiling
#define KC 32                 // K-chunk staged in LDS per iteration
#define BPAD 36               // padded floats per B column (144B = 9 x 16B)

// ---------------------------------------------------------------------------
// GEMM: Dst[m][n] = sum_k A[m][k] * W[n][k] + bias[n]    (A @ W^T)
// Block: 8 waves, tile 128(M) x 64(N); wave owns 16(M) x 64(N) via 4 accs.
// B tile (shared by all 8 waves) is staged in LDS with double-buffered
// global_load_async_to_lds_b128; A fragments are register-prefetched.
// ---------------------------------------------------------------------------
__global__ __launch_bounds__(256)
void cosformer_gemm_wmma(const float* __restrict__ A,
                         const float* __restrict__ W,
                         const float* __restrict__ bias,
                         float* __restrict__ Dst,
                         int M, int N, int K) {
  __shared__ float sB[2][64 * BPAD];

  const int tid = threadIdx.x;
  const int wave = tid >> 5;
  const int lane = tid & 31;
  const int half = lane >> 4;           // k sub-phase
  const int lm = lane & 15;

  const int m0 = blockIdx.y * 128 + wave * 16;
  const int n0 = blockIdx.x * 64;

  const float* __restrict__ aptr = A + (size_t)(m0 + lm) * K + 2 * half;

  // Async-copy mapping: per K-chunk the B tile is 64 cols x 32 floats
  // (copy only first 128B of each padded 144B column row).
  // 512 16-byte segments per stage -> 2 per thread.
  const int s0 = tid, s1 = tid + 256;
  const int col0 = s0 >> 3, off0 = (s0 & 7) * 4;     // offsets in floats
  const int col1 = s1 >> 3, off1 = (s1 & 7) * 4;
  const float* gB0 = W + (size_t)(n0 + col0) * K + off0;
  const float* gB1 = W + (size_t)(n0 + col1) * K + off1;
  const uint32_t l0 = (uint32_t)(uintptr_t)&sB[0][col0 * BPAD + off0];
  const uint32_t l1 = (uint32_t)(uintptr_t)&sB[0][col1 * BPAD + off1];
  const uint32_t bufBytes = (uint32_t)(64 * BPAD * 4);

  auto stage = [&](int buf, int k0) {
    uint64_t g0 = (uint64_t)(uintptr_t)(gB0 + k0);
    uint64_t g1 = (uint64_t)(uintptr_t)(gB1 + k0);
    uint32_t d0 = l0 + (uint32_t)buf * bufBytes;
    uint32_t d1 = l1 + (uint32_t)buf * bufBytes;
    asm volatile("global_load_async_to_lds_b128 %0, %1, off"
                 :: "v"(d0), "v"(g0) : "memory");
    asm volatile("global_load_async_to_lds_b128 %0, %1, off"
                 :: "v"(d1), "v"(g1) : "memory");
  };

  v8f acc[4] = {v8f{}, v8f{}, v8f{}, v8f{}};

  const int nchunks = K / KC;
  stage(0, 0);

  for (int c = 0; c < nchunks; ++c) {
    const int buf = c & 1;
    const int k0 = c * KC;

    if (c + 1 < nchunks) {
      stage(buf ^ 1, k0 + KC);
      asm volatile("s_wait_asynccnt 0x2" ::: "memory");  // chunk c copies done
    } else {
      asm volatile("s_wait_asynccnt 0x0" ::: "memory");
    }
    __syncthreads();   // all waves' copies of buf visible

    // Register-prefetch the 8 A fragments of this chunk (one b64 each).
    v2f a[KC / 4];
#pragma unroll
    for (int s = 0; s < KC / 4; ++s) {
      a[s] = *(const v2f*)(aptr + k0 + 4 * s);
    }

    const float* bb = &sB[buf][0];
#pragma unroll
    for (int s = 0; s < KC / 4; ++s) {
      const int kb = 4 * s + 2 * half;
      v2f b0 = *(const v2f*)(bb + (lm + 0) * BPAD + kb);
      v2f b1 = *(const v2f*)(bb + (lm + 16) * BPAD + kb);
      v2f b2 = *(const v2f*)(bb + (lm + 32) * BPAD + kb);
      v2f b3 = *(const v2f*)(bb + (lm + 48) * BPAD + kb);
      acc[0] = __builtin_amdgcn_wmma_f32_16x16x4_f32(false, a[s], false, b0, (short)0, acc[0], false, false);
      acc[1] = __builtin_amdgcn_wmma_f32_16x16x4_f32(false, a[s], false, b1, (short)0, acc[1], false, false);
      acc[2] = __builtin_amdgcn_wmma_f32_16x16x4_f32(false, a[s], false, b2, (short)0, acc[2], false, false);
      acc[3] = __builtin_amdgcn_wmma_f32_16x16x4_f32(false, a[s], false, b3, (short)0, acc[3], false, false);
    }
    __syncthreads();   // done reading buf before it is overwritten
  }

  // Lane holds D rows m0 + r + 8*half, cols n0 + 16*j + lm.
#pragma unroll
  for (int j = 0; j < 4; ++j) {
    const int col = n0 + 16 * j + lm;
    const float bn = bias[col];
#pragma unroll
    for (int r = 0; r < 8; ++r) {
      Dst[(size_t)(m0 + r + 8 * half) * N + col] = acc[j][r] + bn;
    }
  }
}

// ---------------------------------------------------------------------------
// Pass 2a: chunk totals. Block (g, bh); threads (e = tid>>2, c = tid&3).
// ---------------------------------------------------------------------------
__global__ __launch_bounds__(256)
void cosformer_chunk_sums(const float* __restrict__ qkv,   // [M][3C]
                          float* __restrict__ part) {
  const int g = blockIdx.x;
  const int bh = blockIdx.y;
  const int b = bh / HH;
  const int h = bh % HH;

  const int tid = threadIdx.x;
  const int e = tid >> 2;
  const int c = tid & 3;
  const int dbase = c * 16;

  __shared__ float s_kc[DD], s_ks[DD], s_v[DD];

  float Mc[16], Ms[16], zc[16], zs[16];
#pragma unroll
  for (int i = 0; i < 16; ++i) { Mc[i] = 0.f; Ms[i] = 0.f; zc[i] = 0.f; zs[i] = 0.f; }

  const float ang_step = (float)(M_PI * 0.5 / (double)TT);

  for (int t = g * CHUNK; t < (g + 1) * CHUNK; ++t) {
    const float ang = ang_step * (float)t;
    const float cw = __cosf(ang);
    const float sw = __sinf(ang);

    if (tid < DD) {
      const size_t rowoff = ((size_t)(b * TT + t)) * (3 * CC);
      const float kv = qkv[rowoff + CC + h * DD + tid];
      const float vv = qkv[rowoff + 2 * CC + h * DD + tid];
      const float kr = fmaxf(kv, 0.f);
      s_kc[tid] = kr * cw;
      s_ks[tid] = kr * sw;
      s_v[tid] = vv;
    }
    __syncthreads();

    const float ve = s_v[e];
#pragma unroll
    for (int i = 0; i < 16; ++i) {
      const int d = dbase + i;
      const float kcd = s_kc[d], ksd = s_ks[d];
      Mc[i] = fmaf(kcd, ve, Mc[i]);
      Ms[i] = fmaf(ksd, ve, Ms[i]);
      zc[i] += kcd;
      zs[i] += ksd;
    }
    __syncthreads();
  }

  float* p = part + ((size_t)bh * NCHUNK + g) * PART_STRIDE;
#pragma unroll
  for (int i = 0; i < 16; ++i) {
    const int d = dbase + i;
    p[d * DD + e] = Mc[i];
    p[DD * DD + d * DD + e] = Ms[i];
  }
  if (e == 0) {
#pragma unroll
    for (int i = 0; i < 16; ++i) {
      const int d = dbase + i;
      p[2 * DD * DD + d] = zc[i];
      p[2 * DD * DD + DD + d] = zs[i];
    }
  }
}

// ---------------------------------------------------------------------------
// Pass 2b: chunked scan. Block (g, bh). Seed state from chunk totals < g.
// ---------------------------------------------------------------------------
__global__ __launch_bounds__(256)
void cosformer_scan(const float* __restrict__ qkv,    // [M][3C]
                    const float* __restrict__ part,
                    float* __restrict__ attn) {       // [M][C]
  const int g = blockIdx.x;
  const int bh = blockIdx.y;
  const int b = bh / HH;
  const int h = bh % HH;

  const int tid = threadIdx.x;
  const int e = tid >> 2;
  const int c = tid & 3;
  const int dbase = c * 16;

  __shared__ float s_qc[DD], s_qs[DD], s_kc[DD], s_ks[DD], s_v[DD];

  float Sc[16], Ss[16], zc[16], zs[16];
#pragma unroll
  for (int i = 0; i < 16; ++i) { Sc[i] = 0.f; Ss[i] = 0.f; zc[i] = 0.f; zs[i] = 0.f; }

  for (int gp = 0; gp < g; ++gp) {
    const float* p = part + ((size_t)bh * NCHUNK + gp) * PART_STRIDE;
#pragma unroll
    for (int i = 0; i < 16; ++i) {
      const int d = dbase + i;
      Sc[i] += p[d * DD + e];
      Ss[i] += p[DD * DD + d * DD + e];
      zc[i] += p[2 * DD * DD + d];
      zs[i] += p[2 * DD * DD + DD + d];
    }
  }

  const float ang_step = (float)(M_PI * 0.5 / (double)TT);

  for (int t = g * CHUNK; t < (g + 1) * CHUNK; ++t) {
    const float ang = ang_step * (float)t;
    const float cw = __cosf(ang);
    const float sw = __sinf(ang);

    if (tid < DD) {
      const size_t rowoff = ((size_t)(b * TT + t)) * (3 * CC);
      const float qv = qkv[rowoff + h * DD + tid];
      const float kv = qkv[rowoff + CC + h * DD + tid];
      const float vv = qkv[rowoff + 2 * CC + h * DD + tid];
      const float qr = fmaxf(qv, 0.f);
      const float kr = fmaxf(kv, 0.f);
      s_qc[tid] = qr * cw;
      s_qs[tid] = qr * sw;
      s_kc[tid] = kr * cw;
      s_ks[tid] = kr * sw;
      s_v[tid] = vv;
    }
    __syncthreads();

    const float ve = s_v[e];
    float acc_c = 0.f, acc_s = 0.f, acc_n = 0.f;
#pragma unroll
    for (int i = 0; i < 16; ++i) {
      const int d = dbase + i;
      const float kcd = s_kc[d], ksd = s_ks[d];
      const float qcd = s_qc[d], qsd = s_qs[d];
      Sc[i] = fmaf(kcd, ve, Sc[i]);
      acc_c = fmaf(qcd, Sc[i], acc_c);
      Ss[i] = fmaf(ksd, ve, Ss[i]);
      acc_s = fmaf(qsd, Ss[i], acc_s);
      zc[i] += kcd;
      zs[i] += ksd;
      acc_n = fmaf(qcd, zc[i], acc_n);
      acc_n = fmaf(qsd, zs[i], acc_n);
    }

    float num = acc_c + acc_s;
    num += __shfl_xor(num, 1, 32);
    num += __shfl_xor(num, 2, 32);
    acc_n += __shfl_xor(acc_n, 1, 32);
    acc_n += __shfl_xor(acc_n, 2, 32);

    if (c == 0) {
      attn[((size_t)(b * TT + t)) * CC + h * DD + e] = num / (acc_n + EPS);
    }
    __syncthreads();
  }
}

// ---------------------------------------------------------------------------
// Host-side launch.
// ---------------------------------------------------------------------------
extern "C" void kernel_launch(void* const* d_in, const int* in_sizes, int n_in,
                              void* d_out, int out_size, void* d_ws, size_t ws_size,
                              hipStream_t stream) {
  const float* x     = (const float*)d_in[0];
  const float* w_qkv = (const float*)d_in[1];
  const float* b_qkv = (const float*)d_in[2];
  const float* w_out = (const float*)d_in[3];
  const float* b_out = (const float*)d_in[4];
  float* out = (float*)d_out;

  float* qkv  = (float*)d_ws;                              // 3.1M floats
  float* attn = qkv + (size_t)MM * (3 * CC);               // 1.0M floats
  float* part = attn + (size_t)MM * CC;                    // 1.1M floats

  {
    dim3 grid((3 * CC) / 64, MM / 128);
    cosformer_gemm_wmma<<<grid, 256, 0, stream>>>(x, w_qkv, b_qkv, qkv,
                                                  MM, 3 * CC, CC);
  }
  {
    dim3 grid(NCHUNK, BB * HH);
    cosformer_chunk_sums<<<grid, 256, 0, stream>>>(qkv, part);
  }
  {
    dim3 grid(NCHUNK, BB * HH);
    cosformer_scan<<<grid, 256, 0, stream>>>(qkv, part, attn);
  }
  {
    dim3 grid(CC / 64, MM / 128);
    cosformer_gemm_wmma<<<grid, 256, 0, stream>>>(attn, w_out, b_out, out,
                                                  MM, CC, CC);
  }
}